// SCTCSBLoss_10118942949436
// MI455X (gfx1250) — compile-verified
//
#include <hip/hip_runtime.h>
#include <hip/hip_bf16.h>
#include <math.h>

#define NUM_FEATURES 35
#define NEGV (-1e30f)

typedef __attribute__((ext_vector_type(2)))  float v2f;
typedef __attribute__((ext_vector_type(8)))  float v8f;

// ---------------------------------------------------------------------------
// Kernel 1: one workgroup per (b,f) chain; one thread per extended state s.
// Alpha double-buffered in LDS, one barrier per timestep. Per-timestep logit
// x_t and its 3-class log-softmax denominator lse_t are precomputed into LDS
// so the serial loop is pure LDS + VALU (v_exp_f32 / v_log_f32) work.
// ---------------------------------------------------------------------------
__global__ void sctc_alpha_kernel(
    const float* __restrict__ logits,        // (B,T,F)
    const float* __restrict__ blank_logit,   // (1,)
    const int*   __restrict__ targets,       // (B,S,F)
    const int*   __restrict__ input_lengths, // (B,)
    const int*   __restrict__ target_lengths,// (B,)
    float*       __restrict__ per_bf,        // (B*F,)
    int B, int T, int S)
{
    extern __shared__ float smem[];
    const int nthr = blockDim.x;
    float* xb   = smem;            // [T]   logits x_t for this (b,f)
    float* lseb = smem + T;        // [T]   log(e^x + e^-x + e^bl) (stable)
    float* abuf = smem + 2 * T;    // [2*nthr] alpha double buffer

    const int F  = NUM_FEATURES;
    const int bf = blockIdx.x;
    const int b  = bf / F;
    const int f  = bf % F;
    const int s  = threadIdx.x;
    const int E  = 2 * S + 1;

    const float bl = blank_logit[0];

    // --- preload per-timestep scalars ---
    for (int t = s; t < T; t += nthr) {
        float x = logits[((size_t)b * T + t) * F + f];
        xb[t] = x;
        float m = fmaxf(fmaxf(x, -x), bl);
        lseb[t] = m + __logf(__expf(x - m) + __expf(-x - m) + __expf(bl - m));
    }

    // --- per-state label (ext targets) and skip-transition permission ---
    int  label = 2;          // blank
    bool allowSkip = false;
    if ((s & 1) && s < E) {
        label = targets[((size_t)b * S + (s >> 1)) * F + f];
        if (s >= 3) {
            int prev = targets[((size_t)b * S + (s >> 1) - 1) * F + f];
            allowSkip = (label != prev);
        } else {
            allowSkip = true;   // s==1: a2 is NEG anyway
        }
    }
    const float sgn = (label == 1) ? 1.0f : -1.0f;

    __syncthreads();

    // --- t = 0 init ---
    float lp0 = ((label == 2) ? bl : sgn * xb[0]) - lseb[0];
    float a   = (s <= 1) ? lp0 : NEGV;
    int   p   = 0;
    abuf[s] = a;
    __syncthreads();

    const int len = input_lengths[b];   // uniform per block
    for (int t = 1; t < len; ++t) {
        const float* rd = abuf + p * nthr;
        float a1 = (s >= 1) ? rd[s - 1] : NEGV;
        float a2 = (allowSkip && s >= 2) ? rd[s - 2] : NEGV;
        float m  = fmaxf(fmaxf(a, a1), a2);
        float l3 = m + __logf(__expf(a - m) + __expf(a1 - m) + __expf(a2 - m));
        float lp = ((label == 2) ? bl : sgn * xb[t]) - lseb[t];
        a = l3 + lp;
        abuf[(p ^ 1) * nthr + s] = a;
        __syncthreads();
        p ^= 1;
    }

    if (s == 0) {
        const float* fa = abuf + p * nthr;
        int   L   = target_lengths[b];
        float aL  = fa[2 * L];
        float aL1 = fa[2 * L - 1];
        float m   = fmaxf(aL, aL1);
        float ll  = m + __logf(__expf(aL - m) + __expf(aL1 - m));
        float loss = -ll;
        if (loss > 0.5e30f) loss = 0.0f;      // zero_infinity
        per_bf[bf] = loss / (float)L;         // reduction='mean' per-target-len
    }
}

// ---------------------------------------------------------------------------
// Pad kernel: zero d_ws[n .. paddedN) so the WMMA reduction can use
// unconditional vector loads with uniform control flow.
// ---------------------------------------------------------------------------
__global__ __launch_bounds__(64) void pad_zero_kernel(
    float* __restrict__ buf, int n, int paddedN)
{
    for (int i = n + threadIdx.x; i < paddedN; i += 64)
        buf[i] = 0.0f;
}

// ---------------------------------------------------------------------------
// Kernel 2: single-wave mean reduction via WMMA (sum = rowsum(A x ones)).
// B is all-ones so the A layout is irrelevant for the grand total; each
// column of D carries the 16 row sums, hence total = (sum over 32 lanes of
// sum(c[0..7])) / 16. Input is zero-padded to a multiple of 64, so every
// chunk is one unconditional global_load_b64 per lane and EXEC stays all-1s
// (as required for WMMA) with no exec save/restore churn.
// ---------------------------------------------------------------------------
__global__ __launch_bounds__(32) void reduce_mean_wmma_kernel(
    const float* __restrict__ vals, float* __restrict__ out, int n)
{
    const int lane = threadIdx.x;
    v8f c = {};   // f32 accumulator (16x16 C/D, 8 VGPRs/lane, wave32)

    v2f bones; bones[0] = 1.0f; bones[1] = 1.0f;
    const int nChunks = (n + 63) / 64;              // input padded to 64
    const float* p = vals + lane * 2;
    for (int ch = 0; ch < nChunks; ++ch) {
        v2f aa = *(const v2f*)(p + ch * 64);        // global_load_b64
        c = __builtin_amdgcn_wmma_f32_16x16x4_f32(
                false, aa, false, bones, (short)0, c, false, false);
    }

    float s = c[0] + c[1] + c[2] + c[3] + c[4] + c[5] + c[6] + c[7];
    for (int off = 16; off >= 1; off >>= 1)
        s += __shfl_xor(s, off, 32);
    if (lane == 0)
        out[0] = s / (16.0f * (float)n);      // /16 column redundancy, /n mean
}

// ---------------------------------------------------------------------------
extern "C" void kernel_launch(void* const* d_in, const int* in_sizes, int n_in,
                              void* d_out, int out_size, void* d_ws, size_t ws_size,
                              hipStream_t stream) {
    const float* logits         = (const float*)d_in[0];
    const float* blank_logit    = (const float*)d_in[1];
    const int*   targets        = (const int*)d_in[2];
    const int*   input_lengths  = (const int*)d_in[3];
    const int*   target_lengths = (const int*)d_in[4];
    float*       out            = (float*)d_out;

    const int F = NUM_FEATURES;
    const int B = in_sizes[3];
    const int T = in_sizes[0] / (B * F);
    const int S = in_sizes[2] / (B * F);
    const int E = 2 * S + 1;
    const int nbf = B * F;
    const int paddedN = ((nbf + 63) / 64) * 64;

    float* per_bf = (float*)d_ws;   // paddedN floats of scratch

    int threads = ((E + 31) / 32) * 32;     // one thread per state (E=201 -> 224)
    if (threads > 1024) threads = 1024;
    size_t shbytes = (size_t)(2 * T + 2 * threads) * sizeof(float);

    sctc_alpha_kernel<<<nbf, threads, shbytes, stream>>>(
        logits, blank_logit, targets, input_lengths, target_lengths,
        per_bf, B, T, S);

    pad_zero_kernel<<<1, 64, 0, stream>>>(per_bf, nbf, paddedN);

    reduce_mean_wmma_kernel<<<1, 32, 0, stream>>>(per_bf, out, nbf);
}